// AdaptiveFeaturePooling_29497835389403
// MI455X (gfx1250) — compile-verified
//
#include <hip/hip_runtime.h>
#include <hip/hip_bf16.h>

// ---------------------------------------------------------------------------
// AdaptiveFeaturePooling (two-level ROI-align, sum) for MI455X / gfx1250.
// Memory-bound gather: WMMA not applicable. Uses CDNA5 async global->LDS DMA
// (ASYNCcnt path) to stage each ROI's dense feature footprint in LDS, with
// double-buffered channel tiles so DMA overlaps the LDS bilinear gather.
// ---------------------------------------------------------------------------

#ifndef __has_builtin
#define __has_builtin(x) 0
#endif

#if __has_builtin(__builtin_amdgcn_global_load_async_to_lds_b32)
#define USE_ASYNC_LDS 1
#else
#define USE_ASYNC_LDS 0
#endif

#define AS_GLOBAL __attribute__((address_space(1)))
#define AS_LDS    __attribute__((address_space(3)))

namespace {
constexpr int OUT   = 14;
constexpr int BINS  = OUT * OUT;     // 196
constexpr int CH    = 256;
constexpr int CT    = 8;             // channels per LDS tile
constexpr int NT    = CH / CT;       // 32 tiles
constexpr int H4 = 128, W4 = 128;    // stride-4 level
constexpr int H8 = 64,  W8 = 64;     // stride-8 level
constexpr int FH4 = 32, FW4 = 32;    // max footprint, stride-4
constexpr int FH8 = 20, FW8 = 20;    // max footprint, stride-8
constexpr int BLOCK = 256;

// LDS layout (bytes) -- double-buffered tiles
constexpr int W_TAB_OFF = 0;                               // float [2][196][4][4]
constexpr int W_TAB_SZ  = 2 * BINS * 4 * 4 * 4;            // 25088
constexpr int O_TAB_OFF = W_TAB_OFF + W_TAB_SZ;            // uint  [2][196][4][2]
constexpr int O_TAB_SZ  = 2 * BINS * 4 * 2 * 4;            // 12544
constexpr int T4_ELEMS  = CT * FH4 * FW4;                  // 8192
constexpr int T8_ELEMS  = CT * FH8 * FW8;                  // 3200
constexpr int TILE4_OFF = O_TAB_OFF + O_TAB_SZ;            // float [2][T4_ELEMS]
constexpr int TILE4_SZ  = 2 * T4_ELEMS * 4;                // 65536
constexpr int TILE8_OFF = TILE4_OFF + TILE4_SZ;            // float [2][T8_ELEMS]
constexpr int TILE8_SZ  = 2 * T8_ELEMS * 4;                // 25600
constexpr int SMEM_BYTES = TILE8_OFF + TILE8_SZ;           // 128768
}

static __device__ __forceinline__ void async_g2l_b32(const float* g, float* l) {
#if USE_ASYNC_LDS
    __builtin_amdgcn_global_load_async_to_lds_b32(
        (AS_GLOBAL int*)(AS_GLOBAL void*)g,
        (AS_LDS int*)(AS_LDS void*)l, 0, 0);
#else
    *l = *g;
#endif
}

static __device__ __forceinline__ void wait_async_copies() {
#if USE_ASYNC_LDS
#if __has_builtin(__builtin_amdgcn_s_wait_asynccnt)
    __builtin_amdgcn_s_wait_asynccnt(0);
#else
    asm volatile("s_wait_asynccnt 0" ::: "memory");
#endif
#endif
}

__global__ __launch_bounds__(BLOCK)
void AdaptiveFeaturePooling_29497835389403_kernel(
    const float* __restrict__ f4, const float* __restrict__ f8,
    const float* __restrict__ boxes, const int* __restrict__ bidx,
    float* __restrict__ out)
{
    extern __shared__ char smem[];
    float* w_tab = (float*)(smem + W_TAB_OFF);       // [2][BINS][4][4]
    unsigned* o_tab = (unsigned*)(smem + O_TAB_OFF); // [2][BINS][4][2]
    float* tile4 = (float*)(smem + TILE4_OFF);       // [2][T4_ELEMS]
    float* tile8 = (float*)(smem + TILE8_OFF);       // [2][T8_ELEMS]

    const int r   = blockIdx.x;
    const int tid = threadIdx.x;

    // Scalar (uniform) box parameters.
    const float bx1 = boxes[4 * r + 0];
    const float by1 = boxes[4 * r + 1];
    const float bx2 = boxes[4 * r + 2];
    const float by2 = boxes[4 * r + 3];
    const int   b   = bidx[r];

    // Per-level ROI geometry + footprint (computed identically by all threads).
    float X1[2], Y1[2], binw[2], binh[2], gwf[2], ghf[2], invc[2];
    int ghn[2], gwn[2], xlo[2], ylo[2], fwv[2], fhv[2];
#pragma unroll
    for (int lv = 0; lv < 2; ++lv) {
        const float sc = lv ? 0.125f : 0.25f;
        const int   Hh = lv ? H8 : H4;
        const int   Ww = lv ? W8 : W4;
        const int   FWm = lv ? FW8 : FW4;
        const int   FHm = lv ? FH8 : FH4;
        const float x1s = bx1 * sc, y1s = by1 * sc;
        const float x2s = bx2 * sc, y2s = by2 * sc;
        const float rw = fmaxf(x2s - x1s, 1.0f);
        const float rh = fmaxf(y2s - y1s, 1.0f);
        const float bw = rw * (1.0f / OUT);
        const float bh = rh * (1.0f / OUT);
        const int gwi = (int)ceilf(rw * (1.0f / OUT));
        const int ghi = (int)ceilf(rh * (1.0f / OUT));
        const float gwF = (float)gwi, ghF = (float)ghi;
        X1[lv] = x1s; Y1[lv] = y1s; binw[lv] = bw; binh[lv] = bh;
        gwf[lv] = gwF; ghf[lv] = ghF; gwn[lv] = gwi; ghn[lv] = ghi;
        invc[lv] = 1.0f / (gwF * ghF);
        // Sample coordinate range (pre-clip): [min + 0.5*step, 13*bin + 1.5*step]
        const float minx = x1s + 0.5f * (bw / gwF);
        const float maxx = x1s + 13.0f * bw + 1.5f * (bw / gwF);
        const float miny = y1s + 0.5f * (bh / ghF);
        const float maxy = y1s + 13.0f * bh + 1.5f * (bh / ghF);
        const int xl = (int)floorf(fminf(fmaxf(minx, 0.0f), (float)(Ww - 1)));
        const int xh = min((int)floorf(fminf(fmaxf(maxx, 0.0f), (float)(Ww - 1))) + 1, Ww - 1);
        const int yl = (int)floorf(fminf(fmaxf(miny, 0.0f), (float)(Hh - 1)));
        const int yh = min((int)floorf(fminf(fmaxf(maxy, 0.0f), (float)(Hh - 1))) + 1, Hh - 1);
        xlo[lv] = xl; ylo[lv] = yl;
        fwv[lv] = min(xh - xl + 1, FWm);
        fhv[lv] = min(yh - yl + 1, FHm);
    }

    // ---- Phase 1: per-bin weight/offset tables (tid == bin) ----
    if (tid < BINS) {
        const int py = tid / OUT;
        const int px = tid % OUT;
#pragma unroll
        for (int lv = 0; lv < 2; ++lv) {
            const int Hh = lv ? H8 : H4;
            const int Ww = lv ? W8 : W4;
            const int FWs = lv ? FW8 : FW4;
            const float fpy = (float)py, fpx = (float)px;
#pragma unroll
            for (int sy = 0; sy < 2; ++sy) {
#pragma unroll
                for (int sx = 0; sx < 2; ++sx) {
                    const int s = sy * 2 + sx;
                    const float y = Y1[lv] + fpy * binh[lv] + ((float)sy + 0.5f) * (binh[lv] / ghf[lv]);
                    const float x = X1[lv] + fpx * binw[lv] + ((float)sx + 0.5f) * (binw[lv] / gwf[lv]);
                    const bool valid = (y >= -1.0f) && (y <= (float)Hh) &&
                                       (x >= -1.0f) && (x <= (float)Ww);
                    const bool smask = (sy < ghn[lv]) && (sx < gwn[lv]);
                    const float m = (valid && smask) ? invc[lv] : 0.0f;
                    const float yc = fminf(fmaxf(y, 0.0f), (float)(Hh - 1));
                    const float xc = fminf(fmaxf(x, 0.0f), (float)(Ww - 1));
                    const int y0 = (int)floorf(yc);
                    const int x0 = (int)floorf(xc);
                    const int y1i = min(y0 + 1, Hh - 1);
                    const int x1i = min(x0 + 1, Ww - 1);
                    const float ly = yc - (float)y0, lx = xc - (float)x0;
                    const float hy = 1.0f - ly,      hx = 1.0f - lx;
                    // Footprint-relative tap offsets (clamped for safety).
                    const int ry0 = min(max(y0  - ylo[lv], 0), fhv[lv] - 1);
                    const int ry1 = min(max(y1i - ylo[lv], 0), fhv[lv] - 1);
                    const int rx0 = min(max(x0  - xlo[lv], 0), fwv[lv] - 1);
                    const int rx1 = min(max(x1i - xlo[lv], 0), fwv[lv] - 1);
                    const unsigned o00 = (unsigned)(ry0 * FWs + rx0);
                    const unsigned o01 = (unsigned)(ry0 * FWs + rx1);
                    const unsigned o10 = (unsigned)(ry1 * FWs + rx0);
                    const unsigned o11 = (unsigned)(ry1 * FWs + rx1);
                    const int base = ((lv * BINS + tid) * 4 + s);
                    w_tab[base * 4 + 0] = hy * hx * m;
                    w_tab[base * 4 + 1] = hy * lx * m;
                    w_tab[base * 4 + 2] = ly * hx * m;
                    w_tab[base * 4 + 3] = ly * lx * m;
                    o_tab[base * 2 + 0] = o00 | (o01 << 16);
                    o_tab[base * 2 + 1] = o10 | (o11 << 16);
                }
            }
        }
    }
    __syncthreads();

    // Hoist this bin's weights/offsets into registers (channel-invariant).
    float    wreg[2][4][4];
    unsigned oreg[2][4][2];
    if (tid < BINS) {
#pragma unroll
        for (int lv = 0; lv < 2; ++lv)
#pragma unroll
            for (int s = 0; s < 4; ++s) {
                const int base = ((lv * BINS + tid) * 4 + s);
#pragma unroll
                for (int k = 0; k < 4; ++k) wreg[lv][s][k] = w_tab[base * 4 + k];
                oreg[lv][s][0] = o_tab[base * 2 + 0];
                oreg[lv][s][1] = o_tab[base * 2 + 1];
            }
    }

    const float* gb4 = f4 + (size_t)b * CH * H4 * W4;
    const float* gb8 = f8 + (size_t)b * CH * H8 * W8;
    const int fh4 = fhv[0], fw4 = fwv[0], yl4 = ylo[0], xl4 = xlo[0];
    const int fh8 = fhv[1], fw8 = fwv[1], yl8 = ylo[1], xl8 = xlo[1];

    // Issue the async DMA for one channel tile into LDS buffer `buf`.
    auto issue_tile = [&](int c0, int buf) {
        float* t4 = tile4 + buf * T4_ELEMS;
        float* t8 = tile8 + buf * T8_ELEMS;
        // Stride-4 footprint (power-of-two layout -> shift/mask indexing).
        for (int e = tid; e < T4_ELEMS; e += BLOCK) {
            const int cl = e >> 10;           // / (FH4*FW4)
            const int rem = e & 1023;
            const int fy = rem >> 5;          // / FW4
            const int fx = rem & 31;
            if (fy < fh4 && fx < fw4) {
                const float* g = gb4 + ((size_t)(c0 + cl) * H4 + (yl4 + fy)) * W4 + (xl4 + fx);
                async_g2l_b32(g, t4 + e);
            }
        }
        // Stride-8 footprint (const-divide -> mul/shift).
        for (int e = tid; e < T8_ELEMS; e += BLOCK) {
            const int cl = e / (FH8 * FW8);
            const int rem = e % (FH8 * FW8);
            const int fy = rem / FW8;
            const int fx = rem % FW8;
            if (fy < fh8 && fx < fw8) {
                const float* g = gb8 + ((size_t)(c0 + cl) * H8 + (yl8 + fy)) * W8 + (xl8 + fx);
                async_g2l_b32(g, t8 + e);
            }
        }
    };

    // ---- Phase 2: double-buffered pipeline — DMA tile t+1 overlaps compute t
    issue_tile(0, 0);
    for (int t = 0; t < NT; ++t) {
        wait_async_copies();   // drain this wave's outstanding async copies
        __syncthreads();       // all waves' copies of tile t now visible in LDS
        if (t + 1 < NT) issue_tile((t + 1) * CT, (t + 1) & 1);

        if (tid < BINS) {
            const float* b4 = tile4 + (t & 1) * T4_ELEMS;
            const float* b8 = tile8 + (t & 1) * T8_ELEMS;
            const int c0 = t * CT;
#pragma unroll
            for (int cl = 0; cl < CT; ++cl) {
                const float* p4 = b4 + cl * (FH4 * FW4);
                const float* p8 = b8 + cl * (FH8 * FW8);
                float acc = 0.0f;
#pragma unroll
                for (int s = 0; s < 4; ++s) {
                    const unsigned pa = oreg[0][s][0], pb = oreg[0][s][1];
                    acc = fmaf(wreg[0][s][0], p4[pa & 0xFFFFu], acc);
                    acc = fmaf(wreg[0][s][1], p4[pa >> 16],     acc);
                    acc = fmaf(wreg[0][s][2], p4[pb & 0xFFFFu], acc);
                    acc = fmaf(wreg[0][s][3], p4[pb >> 16],     acc);
                }
#pragma unroll
                for (int s = 0; s < 4; ++s) {
                    const unsigned pa = oreg[1][s][0], pb = oreg[1][s][1];
                    acc = fmaf(wreg[1][s][0], p8[pa & 0xFFFFu], acc);
                    acc = fmaf(wreg[1][s][1], p8[pa >> 16],     acc);
                    acc = fmaf(wreg[1][s][2], p8[pb & 0xFFFFu], acc);
                    acc = fmaf(wreg[1][s][3], p8[pb >> 16],     acc);
                }
                out[((size_t)r * CH + (c0 + cl)) * BINS + tid] = acc;
            }
        }
        // No trailing barrier: buffer (t&1) is next written by copies issued
        // at iteration t+1 (for tile t+2), which sit after the wait+barrier of iteration t+1
        // (i.e. after every thread has finished reading it).
    }
}

extern "C" void kernel_launch(void* const* d_in, const int* in_sizes, int n_in,
                              void* d_out, int out_size, void* d_ws, size_t ws_size,
                              hipStream_t stream) {
    (void)n_in; (void)out_size; (void)d_ws; (void)ws_size;
    const float* f4    = (const float*)d_in[0];
    const float* f8    = (const float*)d_in[1];
    const float* boxes = (const float*)d_in[2];
    const int*   bi    = (const int*)d_in[3];
    float*       outp  = (float*)d_out;
    const int R = in_sizes[2] / 4;   // boxes is [R,4]
    hipLaunchKernelGGL(AdaptiveFeaturePooling_29497835389403_kernel,
                       dim3(R), dim3(BLOCK), SMEM_BYTES, stream,
                       f4, f8, boxes, bi, outp);
}